// TemporalColor_SSM_Multiscale_22084721836736
// MI455X (gfx1250) — compile-verified
//
#include <hip/hip_runtime.h>
#include <hip/hip_bf16.h>
#include <math.h>

// ---------------------------------------------------------------------------
// Problem constants (fixed by the reference module)
// ---------------------------------------------------------------------------
#define D_MODEL   256
#define N_HEADS   8
#define N_LEVELS  4
#define N_POINTS  4
#define DH        (D_MODEL / N_HEADS)      // 32 == wave32 lanes
#define BATCH     8
#define LQ        5440                     // 64*64 + 32*32 + 16*16 + 8*8
#define M_ROWS    (BATCH * LQ)             // 43520 = 340 * 128

// GEMM tiling
#define BLK_M     128                      // 8 waves x 16 rows
#define BLK_N     64                       // per-wave N strip (4 x 16)
#define KCHUNK    128                      // K rows staged in LDS per pass
// LDS B panel: K-pair interleaved, padded row stride.
//   Bs[kpair * BS_STRIDE + n*2 + (k&1)] = B[k][n]
// BS_STRIDE = 160 floats: 160 mod 64 == 32 banks -> the two half-waves
// (which read kpair rows kp and kp+1) hit disjoint 32-bank halves.
#define BS_STRIDE 160
#define BS_FLOATS ((KCHUNK / 2) * BS_STRIDE)   // 64 * 160 = 10240 floats = 40 KB

typedef __attribute__((ext_vector_type(2))) float v2f;
typedef __attribute__((ext_vector_type(8))) float v8f;

// ---------------------------------------------------------------------------
// FP32 WMMA GEMM:  D[M,Nc] = A[M,K] * B[K,Nc] + bias[Nc]
//
// Block = 256 threads (8 waves). Block computes a BLK_M x BLK_N output tile:
//   - B panel staged in LDS, K-pair interleaved so each lane's {B[k][n],
//     B[k+1][n]} WMMA fragment is one aligned ds_load_b64 (no repack movs)
//   - each wave owns a 16 x 64 strip: 4 independent V_WMMA_F32_16X16X4_F32
//     accumulator chains per k-step (good XDL pipelining)
// Requires: M % BLK_M == 0, Nc % BLK_N == 0, K % KCHUNK == 0 (true here).
// ---------------------------------------------------------------------------
__global__ void gemm_wmma_f32(const float* __restrict__ A,
                              const float* __restrict__ B,
                              const float* __restrict__ bias,
                              float* __restrict__ D,
                              int M, int K, int Nc)
{
    __shared__ float Bs[BS_FLOATS];        // 40 KB

    const int stripsN = Nc / BLK_N;
    const int tileM   = blockIdx.x / stripsN;
    const int strip   = blockIdx.x - tileM * stripsN;
    const int n0      = strip * BLK_N;

    const int waveId  = threadIdx.x >> 5;
    const int lane    = threadIdx.x & 31;
    const int l16     = lane & 15;
    const int halfSel = lane >> 4;         // 0: K={0,1}, 1: K={2,3}

    const int rowA = tileM * BLK_M + waveId * 16 + l16;  // A fragment M index
    const float* __restrict__ Arow = A + (size_t)rowA * K;

    // C/D layout: lane(0..15)->N=lane, VGPR r -> M = r (+8 for lanes 16..31).
    // Bias depends only on the column -> broadcast into the accumulators.
    v8f acc[4];
#pragma unroll
    for (int s = 0; s < 4; ++s) {
        const float bcol = bias[n0 + s * 16 + l16];
#pragma unroll
        for (int r = 0; r < 8; ++r) acc[s][r] = bcol;
    }

    for (int kc = 0; kc < K; kc += KCHUNK) {
        __syncthreads();                   // protect previous chunk's readers
        // Cooperative fill, K-pair interleaved:
        //   task i -> K-pair rp (64 of them), column quad cq (16 of them).
        //   Read two B rows' float4, write two interleaved b128 stores.
        for (int i = threadIdx.x; i < (KCHUNK / 2) * (BLK_N / 4); i += blockDim.x) {
            const int rp = i >> 4;                 // K-pair index in chunk
            const int cq = (i & 15) << 2;          // starting column (x4)
            const float4 ra = *(const float4*)(B + (size_t)(kc + 2 * rp)     * Nc + n0 + cq);
            const float4 rb = *(const float4*)(B + (size_t)(kc + 2 * rp + 1) * Nc + n0 + cq);
            float* dst = Bs + rp * BS_STRIDE + cq * 2;
            *(float4*)(dst)     = make_float4(ra.x, rb.x, ra.y, rb.y);
            *(float4*)(dst + 4) = make_float4(ra.z, rb.z, ra.w, rb.w);
        }
        __syncthreads();

#pragma unroll 4
        for (int k0 = 0; k0 < KCHUNK; k0 += 4) {
            const int kp = (k0 >> 1) + halfSel;    // lane's K-pair row
            v2f a;
            a[0] = Arow[kc + 2 * kp];              // == kc + k0 + halfSel*2
            a[1] = Arow[kc + 2 * kp + 1];
#pragma unroll
            for (int s = 0; s < 4; ++s) {
                const int cn = s * 16 + l16;
                const v2f b = *(const v2f*)(Bs + kp * BS_STRIDE + cn * 2);
                acc[s] = __builtin_amdgcn_wmma_f32_16x16x4_f32(
                             /*neg_a=*/false, a, /*neg_b=*/false, b,
                             /*c_mod=*/(short)0, acc[s],
                             /*reuse_a=*/false, /*reuse_b=*/false);
            }
        }
    }

#pragma unroll
    for (int s = 0; s < 4; ++s) {
        const int col = n0 + s * 16 + l16;
#pragma unroll
        for (int r = 0; r < 8; ++r) {
            const int m = tileM * BLK_M + waveId * 16 + r + (halfSel << 3);
            D[(size_t)m * Nc + col] = acc[s][r];
        }
    }
}

// ---------------------------------------------------------------------------
// Softmax over the 16 (level x point) logits per (n, q, head).
// aw layout: (N, Lq, H, 16) contiguous -> rows of 16.
// ---------------------------------------------------------------------------
__global__ void softmax16(float* __restrict__ aw, int rows)
{
    const int r = blockIdx.x * blockDim.x + threadIdx.x;
    if (r >= rows) return;
    float* p = aw + (size_t)r * 16;

    float mx = p[0];
#pragma unroll
    for (int i = 1; i < 16; ++i) mx = fmaxf(mx, p[i]);
    float e[16];
    float s = 0.f;
#pragma unroll
    for (int i = 0; i < 16; ++i) { e[i] = __expf(p[i] - mx); s += e[i]; }
    const float inv = 1.0f / s;
#pragma unroll
    for (int i = 0; i < 16; ++i) p[i] = e[i] * inv;
}

// ---------------------------------------------------------------------------
// Bilinear sampling + attention-weighted accumulation.
// One wave per (n, q, head); lane == channel within dh (32 channels, wave32).
// All sampling coordinates are wave-uniform -> bounds branches don't diverge.
// ---------------------------------------------------------------------------
__global__ void msda_sample(const float* __restrict__ v,
                            const float* __restrict__ off,
                            const float* __restrict__ aw,
                            const float* __restrict__ ref,
                            float* __restrict__ outp,
                            int total /* N*Lq*H */)
{
    const int wavesPerBlock = blockDim.x >> 5;
    const int item = blockIdx.x * wavesPerBlock + (threadIdx.x >> 5);
    if (item >= total) return;
    const int lane = threadIdx.x & 31;

    const int h  = item & (N_HEADS - 1);
    const int nq = item >> 3;                 // n*Lq + q
    const int n  = nq / LQ;

    const int Hs[N_LEVELS] = {64, 32, 16, 8};
    const int Ws[N_LEVELS] = {64, 32, 16, 8};
    const int St[N_LEVELS] = {0, 4096, 5120, 5376};

    const float* __restrict__ vhead =
        v + ((size_t)n * LQ) * D_MODEL + h * DH + lane;
    const float* __restrict__ offp = off + (size_t)item * (N_LEVELS * N_POINTS * 2);
    const float* __restrict__ awp  = aw  + (size_t)item * (N_LEVELS * N_POINTS);
    const float* __restrict__ refp = ref + (size_t)nq * (N_LEVELS * 2);

    float acc = 0.0f;

#pragma unroll
    for (int l = 0; l < N_LEVELS; ++l) {
        const int   Hl = Hs[l], Wl = Ws[l], start = St[l];
        const float rx = refp[l * 2 + 0];
        const float ry = refp[l * 2 + 1];

#pragma unroll
        for (int p = 0; p < N_POINTS; ++p) {
            const float ox = offp[(l * N_POINTS + p) * 2 + 0];
            const float oy = offp[(l * N_POINTS + p) * 2 + 1];
            // loc = ref + off/norm;  x = loc.x*W - 0.5  ->  rx*W + ox - 0.5
            const float x = rx * (float)Wl + ox - 0.5f;
            const float y = ry * (float)Hl + oy - 0.5f;

            const float x0f = floorf(x);
            const float y0f = floorf(y);
            const float wx  = x - x0f;
            const float wy  = y - y0f;
            const int   x0  = (int)x0f;
            const int   y0  = (int)y0f;

            auto corner = [&](int yi, int xi) -> float {
                if (xi < 0 || xi >= Wl || yi < 0 || yi >= Hl) return 0.0f;
                return vhead[(size_t)(start + yi * Wl + xi) * D_MODEL];
            };
            const float v00 = corner(y0,     x0);
            const float v01 = corner(y0,     x0 + 1);
            const float v10 = corner(y0 + 1, x0);
            const float v11 = corner(y0 + 1, x0 + 1);

            const float bil = v00 * (1.f - wx) * (1.f - wy)
                            + v01 * wx        * (1.f - wy)
                            + v10 * (1.f - wx) * wy
                            + v11 * wx        * wy;
            acc = fmaf(awp[l * N_POINTS + p], bil, acc);
        }
    }

    outp[(size_t)nq * D_MODEL + h * DH + lane] = acc;
}

// ---------------------------------------------------------------------------
// Host-side launcher
// ---------------------------------------------------------------------------
extern "C" void kernel_launch(void* const* d_in, const int* in_sizes, int n_in,
                              void* d_out, int out_size, void* d_ws, size_t ws_size,
                              hipStream_t stream)
{
    const float* query   = (const float*)d_in[0];   // (N, Lq, 256)
    const float* refpts  = (const float*)d_in[1];   // (N, Lq, L, 2)
    // d_in[2] spatial_shapes, d_in[3] level_start_index: fixed constants, hardcoded
    const float* W_value = (const float*)d_in[4];
    const float* b_value = (const float*)d_in[5];
    const float* W_off   = (const float*)d_in[6];   // (256, 256)
    const float* b_off   = (const float*)d_in[7];
    const float* W_attn  = (const float*)d_in[8];   // (256, 128)
    const float* b_attn  = (const float*)d_in[9];
    const float* W_out   = (const float*)d_in[10];
    const float* b_out   = (const float*)d_in[11];
    float* out = (float*)d_out;

    // Workspace layout (floats)
    float* ws    = (float*)d_ws;
    float* v     = ws;                                      // M * 256
    float* offb  = v    + (size_t)M_ROWS * D_MODEL;         // M * 256
    float* awb   = offb + (size_t)M_ROWS * D_MODEL;         // M * 128
    float* outs  = awb  + (size_t)M_ROWS * 128;             // M * 256

    const int threads = 256;                 // 8 waves per block
    const int wavesPerBlock = threads / 32;

    // 1) v = query @ W_value + b_value         (43520 x 256 x 256)
    {
        const int blocks = (M_ROWS / BLK_M) * (D_MODEL / BLK_N);
        gemm_wmma_f32<<<blocks, threads, 0, stream>>>(
            query, W_value, b_value, v, M_ROWS, D_MODEL, D_MODEL);
    }
    // 2) off = v @ W_off + b_off               (43520 x 256 x 256)
    {
        const int blocks = (M_ROWS / BLK_M) * (D_MODEL / BLK_N);
        gemm_wmma_f32<<<blocks, threads, 0, stream>>>(
            v, W_off, b_off, offb, M_ROWS, D_MODEL, D_MODEL);
    }
    // 3) attn logits = v @ W_attn + b_attn     (43520 x 256 x 128)
    {
        const int blocks = (M_ROWS / BLK_M) * (128 / BLK_N);
        gemm_wmma_f32<<<blocks, threads, 0, stream>>>(
            v, W_attn, b_attn, awb, M_ROWS, D_MODEL, 128);
    }
    // 4) softmax over 16 logits per (n,q,head)
    {
        const int rows = M_ROWS * N_HEADS;
        const int blocks = (rows + threads - 1) / threads;
        softmax16<<<blocks, threads, 0, stream>>>(awb, rows);
    }
    // 5) bilinear sampling + weighted accumulation
    {
        const int items = M_ROWS * N_HEADS;
        const int blocks = (items + wavesPerBlock - 1) / wavesPerBlock;
        msda_sample<<<blocks, threads, 0, stream>>>(
            v, offb, awb, refpts, outs, items);
    }
    // 6) out = outs @ W_out + b_out            (43520 x 256 x 256)
    {
        const int blocks = (M_ROWS / BLK_M) * (D_MODEL / BLK_N);
        gemm_wmma_f32<<<blocks, threads, 0, stream>>>(
            outs, W_out, b_out, out, M_ROWS, D_MODEL, D_MODEL);
    }
}